// H2G2_Net_45595372814386
// MI455X (gfx1250) — compile-verified
//
#include <hip/hip_runtime.h>

// ---------------- problem constants (fixed by the reference) ----------------
#define N_NODES 50000
#define N_EDGES 800000
#define CH      64          // IN == H == 64
#define NREL    4
#define NCLS    4
#define NGRAPH  512
#define WCOLS   (NREL*CH + CH)   // 320: [W_r0 | W_r1 | W_r2 | W_r3 | root]
#define LDS_PITCH 68             // padded row pitch (floats); 272B, 16B-aligned

typedef float v2f __attribute__((ext_vector_type(2)));
typedef float v8f __attribute__((ext_vector_type(8)));
typedef int   v4i __attribute__((vector_size(16)));     // matches builtin param type

#if defined(__has_builtin)
#  if __has_builtin(__builtin_amdgcn_wmma_f32_16x16x4_f32)
#    define USE_WMMA_F32 1
#  endif
#  if __has_builtin(__builtin_amdgcn_global_load_async_to_lds_b128)
#    define HAVE_ASYNC_LDS 1
#  endif
#  if __has_builtin(__builtin_amdgcn_s_wait_asynccnt)
#    define WAIT_ASYNC() __builtin_amdgcn_s_wait_asynccnt(0)
#  endif
#endif
#ifndef WAIT_ASYNC
#  define WAIT_ASYNC() asm volatile("s_wait_asynccnt 0" ::: "memory")
#endif

// global (AS1) / LDS (AS3) v4i pointer casts: addrspace cast on void*, then
// same-AS pointee cast to the int4 type the async-LDS builtins expect.
#define AS1V4(p) ((__attribute__((address_space(1))) v4i*)                      \
                  (__attribute__((address_space(1))) void*)(void*)(p))
#define AS3V4(p) ((__attribute__((address_space(3))) v4i*)                      \
                  (__attribute__((address_space(3))) void*)(p))

// ---------------- small utility kernels ----------------
__global__ void zero_f32(float* __restrict__ p, int n) {
    int i = blockIdx.x * blockDim.x + threadIdx.x;
    if (i < n) p[i] = 0.0f;
}

// Wcat[k][j]: j<256 -> (sum_b comp[r,b]*basis[b])[k][j%64], r=j/64 ; j>=256 -> root[k][j-256]
__global__ void build_wcat(const float* __restrict__ basis,  // [R][CH][CH]
                           const float* __restrict__ comp,   // [R][R]
                           const float* __restrict__ root,   // [CH][CH]
                           float* __restrict__ wcat)         // [CH][WCOLS]
{
    int idx = blockIdx.x * blockDim.x + threadIdx.x;
    if (idx >= CH * WCOLS) return;
    int k = idx / WCOLS;
    int j = idx % WCOLS;
    float v;
    if (j < NREL * CH) {
        int r = j >> 6, col = j & 63;
        v = 0.0f;
        #pragma unroll
        for (int b = 0; b < NREL; ++b)
            v += comp[r * NREL + b] * basis[((size_t)b * CH + k) * CH + col];
    } else {
        v = root[k * CH + (j - NREL * CH)];
    }
    wcat[idx] = v;
}

// per-(relation,node) in-degree + per-graph node counts
__global__ void deg_count(const int* __restrict__ edge_index,  // [2][E]
                          const int* __restrict__ edge_type,   // [E]
                          const int* __restrict__ batch,       // [N]
                          float* __restrict__ deg,             // [R][N]
                          float* __restrict__ cntg)            // [G]
{
    int i = blockIdx.x * blockDim.x + threadIdx.x;
    if (i < N_EDGES) {
        int t = edge_type[i];
        int d = edge_index[N_EDGES + i];
        atomicAdd(&deg[(size_t)t * N_NODES + d], 1.0f);
    } else if (i < N_EDGES + N_NODES) {
        int n = i - N_EDGES;
        atomicAdd(&cntg[batch[n]], 1.0f);
    }
}

__global__ void recip_deg(float* __restrict__ deg, int n) {  // deg -> 1/max(deg,1)
    int i = blockIdx.x * blockDim.x + threadIdx.x;
    if (i < n) deg[i] = 1.0f / fmaxf(deg[i], 1.0f);
}

// ---------------- dense transform ----------------
// One wave computes a 16(M) x 64(N) strip: 4 accumulators, 64 f32-WMMAs.
// Column group g: g<4 -> hrel[g][n][0:64] ; g==4 -> acc[n][0:64] = self + bias.
// A tile and Wcat panel staged in LDS via async global->LDS loads.
template <int RELU_A>
__global__ __launch_bounds__(32)
void rgcn_gemm_wmma(const float* __restrict__ A,
                    const float* __restrict__ Wcat,
                    const float* __restrict__ bias,
                    float* __restrict__ hrel,   // [NREL][N_NODES][CH]
                    float* __restrict__ acc)    // [N_NODES][CH]
{
    const int g    = blockIdx.x;       // 0..4 (column group of 64)
    const int mt   = blockIdx.y;       // 0..3124  (50000 = 16*3125 exactly)
    const int lane = threadIdx.x;      // wave32
    const int half = lane >> 4;
    const int lrow = lane & 15;
    const int m0 = mt * 16;
    const int c0 = g * 64;

    __shared__ float sA[16 * LDS_PITCH];   // 16 rows x 64 K (padded)
    __shared__ float sB[CH * LDS_PITCH];   // 64 K   x 64 cols (padded)

#if HAVE_ASYNC_LDS
    // A tile: 1024 floats = 8 issues of (32 lanes x 16B)
    #pragma unroll
    for (int i = 0; i < 8; ++i) {
        int chunk = i * 32 + lane;               // 0..255
        int row = chunk >> 4;                    // 0..15
        int col = (chunk & 15) * 4;              // 0,4,..,60
        __builtin_amdgcn_global_load_async_to_lds_b128(
            AS1V4(A + (size_t)(m0 + row) * CH + col),
            AS3V4(sA + row * LDS_PITCH + col), 0, 0);
    }
    // B panel: 4096 floats = 32 issues
    #pragma unroll
    for (int i = 0; i < 32; ++i) {
        int chunk = i * 32 + lane;               // 0..1023
        int row = chunk >> 4;                    // 0..63 (K)
        int col = (chunk & 15) * 4;
        __builtin_amdgcn_global_load_async_to_lds_b128(
            AS1V4(Wcat + (size_t)row * WCOLS + c0 + col),
            AS3V4(sB + row * LDS_PITCH + col), 0, 0);
    }
    WAIT_ASYNC();
#else
    for (int idx = lane; idx < 16 * CH; idx += 32) {
        int row = idx >> 6, col = idx & 63;
        sA[row * LDS_PITCH + col] = A[(size_t)(m0 + row) * CH + col];
    }
    for (int idx = lane; idx < CH * 64; idx += 32) {
        int row = idx >> 6, col = idx & 63;
        sB[row * LDS_PITCH + col] = Wcat[(size_t)row * WCOLS + c0 + col];
    }
#endif

    v8f c[4] = {};

#if USE_WMMA_F32
    #pragma unroll
    for (int kk = 0; kk < 16; ++kk) {
        const int kb = kk * 4 + 2 * half;   // A vgpr0: K=kb (lanes0-15) / kb+2 (lanes16-31)
        v2f a;
        float a0 = sA[lrow * LDS_PITCH + kb];
        float a1 = sA[lrow * LDS_PITCH + kb + 1];
        if (RELU_A) { a0 = fmaxf(a0, 0.0f); a1 = fmaxf(a1, 0.0f); }
        a[0] = a0; a[1] = a1;
        #pragma unroll
        for (int t = 0; t < 4; ++t) {
            v2f b;
            b[0] = sB[kb * LDS_PITCH + t * 16 + lrow];
            b[1] = sB[(kb + 1) * LDS_PITCH + t * 16 + lrow];
            c[t] = __builtin_amdgcn_wmma_f32_16x16x4_f32(false, a, false, b,
                                                         (short)0, c[t], false, false);
        }
    }
#else
    // Scalar fallback — compiled only for the host stub (where __has_builtin
    // can't see amdgcn builtins) or a non-WMMA device. Same math as above.
    #pragma unroll
    for (int t = 0; t < 4; ++t) {
        #pragma unroll
        for (int v = 0; v < 8; ++v) {
            const int m = half * 8 + v;
            const int col = t * 16 + lrow;
            float s = 0.0f;
            for (int k = 0; k < CH; ++k) {
                float a = sA[m * LDS_PITCH + k];
                if (RELU_A) a = fmaxf(a, 0.0f);
                s += a * sB[k * LDS_PITCH + col];
            }
            c[t][v] = s;
        }
    }
#endif

    // C/D layout: vgpr v holds row M = 8*half + v, col N = lrow
    #pragma unroll
    for (int t = 0; t < 4; ++t) {
        const int n0 = c0 + t * 16;
        if (n0 < NREL * CH) {
            const int col = (n0 & 63) + lrow;
            float* dst = hrel + (size_t)g * N_NODES * CH;   // g == relation
            #pragma unroll
            for (int v = 0; v < 8; ++v)
                dst[(size_t)(m0 + half * 8 + v) * CH + col] = c[t][v];
        } else {
            const int col = (n0 - NREL * CH) + lrow;
            const float bv = bias[col];
            #pragma unroll
            for (int v = 0; v < 8; ++v)
                acc[(size_t)(m0 + half * 8 + v) * CH + col] = c[t][v] + bv;
        }
    }
}

// ---------------- edge scatter: per-relation mean, pre-scaled by 1/deg ----------------
__global__ void scatter_mean(const int* __restrict__ edge_index,
                             const int* __restrict__ edge_type,
                             const float* __restrict__ hrel,     // [R][N][CH]
                             const float* __restrict__ invdeg,   // [R][N]
                             float* __restrict__ acc)            // [N][CH]
{
    long long gid = (long long)blockIdx.x * blockDim.x + threadIdx.x;
    if (gid >= (long long)N_EDGES * CH) return;
    int e = (int)(gid >> 6);
    int cc = (int)(gid & 63);
    int t = edge_type[e];
    int s = edge_index[e];
    int d = edge_index[N_EDGES + e];
    float v = hrel[((size_t)t * N_NODES + s) * CH + cc] * invdeg[(size_t)t * N_NODES + d];
    atomicAdd(&acc[(size_t)d * CH + cc], v);
}

// ---------------- global mean pool (ReLU fused; batch is sorted) ----------------
// Each thread handles 8 consecutive nodes for one channel; register-accumulate per
// graph run and flush one atomic per run (~8x fewer atomics than naive).
#define POOL_NPT 8
__global__ void pool_sum(const float* __restrict__ acc,   // [N][CH]
                         const int* __restrict__ batch,
                         float* __restrict__ poolsum)     // [G][CH]
{
    int gid = blockIdx.x * blockDim.x + threadIdx.x;
    if (gid >= (N_NODES / POOL_NPT) * CH) return;        // 50000/8 = 6250 exactly
    int chunk = gid >> 6, cc = gid & 63;
    int n0 = chunk * POOL_NPT;
    int cur = batch[n0];
    float s = 0.0f;
    #pragma unroll
    for (int i = 0; i < POOL_NPT; ++i) {
        int n = n0 + i;
        int b = batch[n];
        float v = fmaxf(acc[(size_t)n * CH + cc], 0.0f);
        if (b != cur) {
            atomicAdd(&poolsum[(size_t)cur * CH + cc], s);
            s = 0.0f; cur = b;
        }
        s += v;
    }
    atomicAdd(&poolsum[(size_t)cur * CH + cc], s);
}

// ---------------- classifier: [G][CH] -> [G][C] ----------------
__global__ void classify(const float* __restrict__ poolsum,
                         const float* __restrict__ cntg,
                         const float* __restrict__ clas_w,  // [CH][C]
                         const float* __restrict__ clas_b,  // [C]
                         float* __restrict__ out)           // [G][C]
{
    int gid = blockIdx.x * blockDim.x + threadIdx.x;
    if (gid >= NGRAPH * NCLS) return;
    int g = gid / NCLS, cls = gid % NCLS;
    float inv = 1.0f / fmaxf(cntg[g], 1.0f);
    float s = clas_b[cls];
    #pragma unroll 8
    for (int cc = 0; cc < CH; ++cc)
        s += poolsum[(size_t)g * CH + cc] * inv * clas_w[cc * NCLS + cls];
    out[gid] = s;
}

// ---------------- host-side orchestration ----------------
extern "C" void kernel_launch(void* const* d_in, const int* in_sizes, int n_in,
                              void* d_out, int out_size, void* d_ws, size_t ws_size,
                              hipStream_t stream) {
    (void)in_sizes; (void)n_in; (void)out_size; (void)ws_size;
    const float* x          = (const float*)d_in[0];
    const int*   edge_index = (const int*)  d_in[1];
    const int*   edge_type  = (const int*)  d_in[2];
    const int*   batch      = (const int*)  d_in[3];
    const float* basis1     = (const float*)d_in[4];
    const float* comp1      = (const float*)d_in[5];
    const float* root1      = (const float*)d_in[6];
    const float* bias1      = (const float*)d_in[7];
    const float* basis2     = (const float*)d_in[8];
    const float* comp2      = (const float*)d_in[9];
    const float* root2      = (const float*)d_in[10];
    const float* bias2      = (const float*)d_in[11];
    const float* clas_w     = (const float*)d_in[12];
    const float* clas_b     = (const float*)d_in[13];
    float* out = (float*)d_out;
    float* ws  = (float*)d_ws;

    // workspace layout (floats), total ~19.47M floats (~78 MB) — fits in the 192MB L2
    const size_t off_wc1  = 0;
    const size_t off_wc2  = off_wc1  + (size_t)CH * WCOLS;
    const size_t off_deg  = off_wc2  + (size_t)CH * WCOLS;
    const size_t off_cntg = off_deg  + (size_t)NREL * N_NODES;
    const size_t off_pool = off_cntg + NGRAPH;
    const size_t off_hrel = off_pool + (size_t)NGRAPH * CH;
    const size_t off_acc1 = off_hrel + (size_t)NREL * N_NODES * CH;
    const size_t off_acc2 = off_acc1 + (size_t)N_NODES * CH;

    float* wc1  = ws + off_wc1;
    float* wc2  = ws + off_wc2;
    float* deg  = ws + off_deg;    // becomes invdeg after recip_deg
    float* cntg = ws + off_cntg;
    float* pool = ws + off_pool;
    float* hrel = ws + off_hrel;
    float* acc1 = ws + off_acc1;
    float* acc2 = ws + off_acc2;

    const int T = 256;
    // 1) zero deg + cntg + poolsum (contiguous region)
    {
        int n = NREL * N_NODES + NGRAPH + NGRAPH * CH;
        zero_f32<<<(n + T - 1) / T, T, 0, stream>>>(deg, n);
    }
    // 2) concatenated weights for both layers
    build_wcat<<<(CH * WCOLS + T - 1) / T, T, 0, stream>>>(basis1, comp1, root1, wc1);
    build_wcat<<<(CH * WCOLS + T - 1) / T, T, 0, stream>>>(basis2, comp2, root2, wc2);
    // 3) degrees + per-graph counts, then reciprocal
    deg_count<<<(N_EDGES + N_NODES + T - 1) / T, T, 0, stream>>>(edge_index, edge_type,
                                                                 batch, deg, cntg);
    recip_deg<<<(NREL * N_NODES + T - 1) / T, T, 0, stream>>>(deg, NREL * N_NODES);

    dim3 ggrid(WCOLS / 64, N_NODES / 16);   // (5, 3125)
    // 4) layer 1: WMMA transform then edge scatter
    rgcn_gemm_wmma<0><<<ggrid, 32, 0, stream>>>(x, wc1, bias1, hrel, acc1);
    {
        long long n = (long long)N_EDGES * CH;
        scatter_mean<<<(unsigned)((n + T - 1) / T), T, 0, stream>>>(edge_index, edge_type,
                                                                    hrel, deg, acc1);
    }
    // 5) layer 2 (ReLU fused into A loads at compile time)
    rgcn_gemm_wmma<1><<<ggrid, 32, 0, stream>>>(acc1, wc2, bias2, hrel, acc2);
    {
        long long n = (long long)N_EDGES * CH;
        scatter_mean<<<(unsigned)((n + T - 1) / T), T, 0, stream>>>(edge_index, edge_type,
                                                                    hrel, deg, acc2);
    }
    // 6) global mean pool (ReLU fused) + classifier
    pool_sum<<<((N_NODES / POOL_NPT) * CH + T - 1) / T, T, 0, stream>>>(acc2, batch, pool);
    classify<<<(NGRAPH * NCLS + T - 1) / T, T, 0, stream>>>(pool, cntg, clas_w, clas_b, out);
}